// SmoothTopKGate_54760833024086
// MI455X (gfx1250) — compile-verified
//
#include <hip/hip_runtime.h>

// SmoothTopKGate for MI455X (gfx1250, wave32).
// One lane owns one row of 8 floats entirely in registers:
//   - 2x nontemporal b128 loads (32B/lane, 1KB/wave, coalesced, streams past L2;
//     working set 256MB > 192MB L2, pure streaming)
//   - branchless top-3 selection (v_max_num/v_min_num) for the Newton init
//   - Newton solve on f(theta) = sum sigmoid((v-theta)/tau) - k using
//     v_tanh_f32 (new CDNA5 TRANS op): sigma(x) = (1+tanh(x/2))/2,
//     sigma' terms via sum(t) and sum(t^2) accumulators:
//       f = 0.5*sum(t) + (DIM/2 - k)
//       g = sum sigma(1-sigma) = DIM/4 - 0.25*sum(t^2)
//     -> inner loop per element: 1 sub + 1 tanh + 1 add + 1 fma.
//   - the tanh values of the breaking iteration ARE the outputs (theta is not
//     updated after the convergence check), so the final gates are 8 fmas,
//     no extra transcendentals.

typedef float vf4 __attribute__((ext_vector_type(4)));

#define SK_DIM      8
#define SK_K        2
#define SK_TAU      0.01f
#define SK_INV_TAU  100.0f
#define SK_MAX_ITER 100
#define SK_TOL      1e-3f

// ---- fast device math -------------------------------------------------------
__device__ __forceinline__ float fast_rcp(float x)  { return __builtin_amdgcn_rcpf(x); }
__device__ __forceinline__ float fast_exp2(float x) { return __builtin_amdgcn_exp2f(x); }

#if __has_builtin(__builtin_amdgcn_tanhf)
  #define SK_HAS_TANH 1
  __device__ __forceinline__ float fast_tanh(float x) { return __builtin_amdgcn_tanhf(x); }
#elif __has_builtin(__builtin_amdgcn_tanh_f32)
  #define SK_HAS_TANH 1
  __device__ __forceinline__ float fast_tanh(float x) { return __builtin_amdgcn_tanh_f32(x); }
#else
  #define SK_HAS_TANH 0
#endif

// ---- kernel -----------------------------------------------------------------
__global__ void __launch_bounds__(256)
smooth_topk_gate_kernel(const float* __restrict__ s,
                        float* __restrict__ out,
                        int nrows) {
    int row = blockIdx.x * blockDim.x + threadIdx.x;
    if (row >= nrows) return;

    const vf4* __restrict__ s4 = reinterpret_cast<const vf4*>(s);
    vf4* __restrict__ o4       = reinterpret_cast<vf4*>(out);

    size_t base = (size_t)row * 2;
    vf4 lo = __builtin_nontemporal_load(&s4[base]);
    vf4 hi = __builtin_nontemporal_load(&s4[base + 1]);

    float v[SK_DIM] = { lo.x, lo.y, lo.z, lo.w, hi.x, hi.y, hi.z, hi.w };

    // ---- branchless top-3 (m1 >= m2 >= m3); theta0 = 3rd largest -----------
    float m1 = -__builtin_inff(), m2 = m1, m3 = m1;
    #pragma unroll
    for (int i = 0; i < SK_DIM; ++i) {
        float x  = v[i];
        float b1 = fminf(m1, x);  m1 = fmaxf(m1, x);
        float b2 = fminf(m2, b1); m2 = fmaxf(m2, b1);
        m3 = fmaxf(m3, b2);
    }
    float theta = m3;  // sorted_s[-k-1] with k=2

#if SK_HAS_TANH
    // ---- Newton with v_tanh_f32 --------------------------------------------
    const float cs = 0.5f * SK_INV_TAU;           // 1/(2*tau)
    float u[SK_DIM];
    #pragma unroll
    for (int i = 0; i < SK_DIM; ++i) u[i] = v[i] * cs;

    float t[SK_DIM];
    bool converged = false;
    for (int it = 0; it < SK_MAX_ITER; ++it) {
        float th = theta * cs;
        float st = 0.0f, q = 0.0f;
        #pragma unroll
        for (int i = 0; i < SK_DIM; ++i) {
            float ti = fast_tanh(u[i] - th);      // v_tanh_f32
            t[i] = ti;
            st += ti;
            q = __builtin_fmaf(ti, ti, q);
        }
        // f = sum(m) - k = DIM/2 + st/2 - k ; g = sum m(1-m) = DIM/4 - q/4
        float f = __builtin_fmaf(0.5f, st, (float)SK_DIM * 0.5f - (float)SK_K);
        if (f < SK_TOL) { converged = true; break; }   // break BEFORE update
        float g = __builtin_fmaf(-0.25f, q, (float)SK_DIM * 0.25f);
        // theta - f/df = theta + tau * f / g
        theta = __builtin_fmaf(SK_TAU * f, fast_rcp(g), theta);
    }
    if (!converged) {                              // rare: MAX_ITER exhausted
        float th = theta * cs;
        #pragma unroll
        for (int i = 0; i < SK_DIM; ++i) t[i] = fast_tanh(u[i] - th);
    }

    // gates = (1 + t)/2 from the saved tanh values — no extra trans ops
    vf4 r0, r1;
    r0.x = __builtin_fmaf(0.5f, t[0], 0.5f);
    r0.y = __builtin_fmaf(0.5f, t[1], 0.5f);
    r0.z = __builtin_fmaf(0.5f, t[2], 0.5f);
    r0.w = __builtin_fmaf(0.5f, t[3], 0.5f);
    r1.x = __builtin_fmaf(0.5f, t[4], 0.5f);
    r1.y = __builtin_fmaf(0.5f, t[5], 0.5f);
    r1.z = __builtin_fmaf(0.5f, t[6], 0.5f);
    r1.w = __builtin_fmaf(0.5f, t[7], 0.5f);
#else
    // ---- fallback: exp2 + rcp sigmoid --------------------------------------
    const float c = SK_INV_TAU * 1.4426950408889634f;  // log2(e)/tau
    float m[SK_DIM];
    bool converged = false;
    for (int it = 0; it < SK_MAX_ITER; ++it) {
        float f = -(float)SK_K, g = 0.0f;
        #pragma unroll
        for (int i = 0; i < SK_DIM; ++i) {
            float e  = fast_exp2((theta - v[i]) * c);
            float mi = fast_rcp(1.0f + e);
            m[i] = mi;
            f += mi;
            g = __builtin_fmaf(-mi, mi, g + mi);       // g += mi*(1-mi)
        }
        if (f < SK_TOL) { converged = true; break; }
        theta = __builtin_fmaf(SK_TAU * f, fast_rcp(g), theta);
    }
    if (!converged) {
        #pragma unroll
        for (int i = 0; i < SK_DIM; ++i) {
            float e = fast_exp2((theta - v[i]) * c);
            m[i] = fast_rcp(1.0f + e);
        }
    }
    vf4 r0, r1;
    r0.x = m[0]; r0.y = m[1]; r0.z = m[2]; r0.w = m[3];
    r1.x = m[4]; r1.y = m[5]; r1.z = m[6]; r1.w = m[7];
#endif

    __builtin_nontemporal_store(r0, &o4[base]);
    __builtin_nontemporal_store(r1, &o4[base + 1]);
}

// ---- launch -----------------------------------------------------------------
extern "C" void kernel_launch(void* const* d_in, const int* in_sizes, int n_in,
                              void* d_out, int out_size, void* d_ws, size_t ws_size,
                              hipStream_t stream) {
    (void)n_in; (void)out_size; (void)d_ws; (void)ws_size;
    const float* s = (const float*)d_in[0];
    float* out     = (float*)d_out;
    int nrows      = in_sizes[0] / SK_DIM;   // 4,000,000

    const int block = 256;                   // 8 wave32 waves per block
    const int grid  = (nrows + block - 1) / block;
    smooth_topk_gate_kernel<<<grid, block, 0, stream>>>(s, out, nrows);
}